// DCNv2Pooling_1726576856683
// MI455X (gfx1250) — compile-verified
//
#include <hip/hip_runtime.h>
#include <hip/hip_bf16.h>

// Deformable PSROI pooling (DCNv2) for MI455X / gfx1250.
// fp32 WMMA 16x16x4 with block-diagonal A over 16 bins per workgroup.
// Optional NCHW->NHWC pre-transpose (uses d_ws) so B-matrix gathers are
// contiguous across lanes; SADDR-form 32-bit-offset gathers either way.

typedef __attribute__((ext_vector_type(2))) float v2f;
typedef __attribute__((ext_vector_type(8))) float v8f;

#define SCALEF    0.0625f
#define TRANS_STD 0.1f
#define POOLED    7
#define NBINS     (POOLED*POOLED)   // 49
#define CH        256
#define IH        128
#define IW        128
#define HWSZ      (IH*IW)

// ---------------------------------------------------------------------------
// NCHW (B,C,H,W) -> NHWC (B,H*W,C) tiled transpose (32x32 tiles via LDS).
// ---------------------------------------------------------------------------
__global__ __launch_bounds__(256) void nchw_to_nhwc(
    const float* __restrict__ inp, float* __restrict__ ws)
{
  __shared__ float t[32][33];
  const int hw0 = blockIdx.x * 32;
  const int c0  = blockIdx.y * 32;
  const int b   = blockIdx.z;
  const int tx  = threadIdx.x & 31;
  const int ty  = threadIdx.x >> 5;          // 0..7
  #pragma unroll
  for (int k = 0; k < 4; ++k) {
    const int ci = ty + k * 8;
    t[ci][tx] = inp[(b * CH + (c0 + ci)) * HWSZ + (hw0 + tx)];
  }
  __syncthreads();
  #pragma unroll
  for (int k = 0; k < 4; ++k) {
    const int hi = ty + k * 8;
    ws[((size_t)b * HWSZ + (hw0 + hi)) * CH + (c0 + tx)] = t[tx][hi];
  }
}

// ---------------------------------------------------------------------------
// Main kernel. PIXMUL/CHMUL give element strides of a pixel / a channel:
//   NCHW path: PIXMUL=1,  CHMUL=HWSZ   (feat[c*HWSZ + pix])
//   NHWC path: PIXMUL=CH, CHMUL=1      (feat[pix*CH  + c ])
// ---------------------------------------------------------------------------
template <int PIXMUL, int CHMUL>
__global__ __launch_bounds__(256) void dcn_psroi_wmma(
    const float* __restrict__ feat,    // layout per template
    const float* __restrict__ rois,    // (N, 5)
    const float* __restrict__ offs,    // (N, 2, 7, 7)
    float* __restrict__ out)           // (N, C, 7, 7)
{
  // Per K-chunk record: [0..3] bilinear weights, [4..7] pixel BYTE offsets.
  __shared__ unsigned sChunk[256][8];
  __shared__ float    sVal[16][16];
  __shared__ float    sCnt[16];
  __shared__ int      sBase;           // batch base, elements

  const int n     = blockIdx.y;
  const int mtile = blockIdx.x;
  const int tid   = threadIdx.x;

  // ---------------- Phase 1: one thread per (bin-in-tile, sample) ----------
  {
    const int m   = tid >> 4;
    const int s   = tid & 15;
    const int bin = mtile * 16 + m;

    const float r0 = rois[n*5+0];
    const float x1 = rintf(rois[n*5+1]) * SCALEF - 0.5f;
    const float y1 = rintf(rois[n*5+2]) * SCALEF - 0.5f;
    const float x2 = (rintf(rois[n*5+3]) + 1.0f) * SCALEF - 0.5f;
    const float y2 = (rintf(rois[n*5+4]) + 1.0f) * SCALEF - 0.5f;
    const float roi_w = fmaxf(x2 - x1, 0.1f);
    const float roi_h = fmaxf(y2 - y1, 0.1f);
    const float bin_w = roi_w * (1.0f/POOLED);
    const float bin_h = roi_h * (1.0f/POOLED);
    const float sub_w = bin_w * 0.25f;
    const float sub_h = bin_h * 0.25f;

    if (tid == 0) sBase = ((int)r0) * (CH*HWSZ);

    float w00=0.f, w01=0.f, w10=0.f, w11=0.f, vflag=0.f;
    unsigned p00=0, p01=0, p10=0, p11=0;
    if (bin < NBINS) {
      const int ph = bin / POOLED;
      const int pw = bin % POOLED;
      const float tx = offs[((n*2+0)*POOLED + ph)*POOLED + pw] * TRANS_STD;
      const float ty = offs[((n*2+1)*POOLED + ph)*POOLED + pw] * TRANS_STD;
      const float wstart = (float)pw * bin_w + x1 + tx * roi_w;
      const float hstart = (float)ph * bin_h + y1 + ty * roi_h;
      const int ix = s & 3;
      const int iy = s >> 2;
      float w = wstart + (float)ix * sub_w;
      float h = hstart + (float)iy * sub_h;
      const bool valid = (w >= -0.5f) && (w <= (float)IW - 0.5f) &&
                         (h >= -0.5f) && (h <= (float)IH - 0.5f);
      w = fminf(fmaxf(w, 0.0f), (float)(IW-1));
      h = fminf(fmaxf(h, 0.0f), (float)(IH-1));
      const float xlo = floorf(w), ylo = floorf(h);
      const float dx = w - xlo,   dy = h - ylo;
      const int xl = (int)xlo,        yl = (int)ylo;
      const int xh = (int)ceilf(w),   yh = (int)ceilf(h);
      if (valid) {
        w00 = (1.f-dx)*(1.f-dy);   // (yl, xl)
        w01 = (1.f-dx)*dy;         // (yh, xl)
        w10 = dx*(1.f-dy);         // (yl, xh)
        w11 = dx*dy;               // (yh, xh)
        vflag = 1.f;
      }
      p00 = (unsigned)(yl*IW + xl) * (PIXMUL*4u);
      p01 = (unsigned)(yh*IW + xl) * (PIXMUL*4u);
      p10 = (unsigned)(yl*IW + xh) * (PIXMUL*4u);
      p11 = (unsigned)(yh*IW + xh) * (PIXMUL*4u);
    }
    sChunk[tid][0] = __float_as_uint(w00);
    sChunk[tid][1] = __float_as_uint(w01);
    sChunk[tid][2] = __float_as_uint(w10);
    sChunk[tid][3] = __float_as_uint(w11);
    sChunk[tid][4] = p00;
    sChunk[tid][5] = p01;
    sChunk[tid][6] = p10;
    sChunk[tid][7] = p11;
    sVal[m][s] = vflag;
  }
  __syncthreads();
  if (tid < 16) {
    float c = 0.f;
    #pragma unroll
    for (int s = 0; s < 16; ++s) c += sVal[tid][s];
    sCnt[tid] = c;
  }
  __syncthreads();

  // ---------------- Phase 2: WMMA accumulation -----------------------------
  const int lane   = tid & 31;
  const int wv     = tid >> 5;               // 0..7
  const int lanelo = lane & 15;
  const int k0     = (lane >= 16) ? 2 : 0;   // mirrored A/B lane-half K pairs

  // Wave-uniform scalar base pointer -> SADDR-form loads with u32 offsets.
  const int baseElem = __builtin_amdgcn_readfirstlane(sBase);
  const char* bp = (const char*)feat + (size_t)(unsigned)(baseElem * 4);

  const unsigned cba = (unsigned)((wv     ) * 16 + lanelo) * (CHMUL * 4u);
  const unsigned cbb = (unsigned)((wv + 8 ) * 16 + lanelo) * (CHMUL * 4u);

  v8f acc0 = {};
  v8f acc1 = {};
  for (int mb = 0; mb < 16; ++mb) {
    const bool sel = (lanelo == mb);         // hoisted A-row select
    for (int s = 0; s < 16; ++s) {
      const unsigned* cp = &sChunk[mb * 16 + s][0];
      const float    w0 = __uint_as_float(cp[k0]);
      const float    w1 = __uint_as_float(cp[k0 + 1]);
      const unsigned p0 = cp[4 + k0];
      const unsigned p1 = cp[5 + k0];
      v2f A;
      A.x = sel ? w0 : 0.f;
      A.y = sel ? w1 : 0.f;
      v2f B0, B1;
      B0.x = *(const float*)(bp + (cba + p0));
      B0.y = *(const float*)(bp + (cba + p1));
      B1.x = *(const float*)(bp + (cbb + p0));
      B1.y = *(const float*)(bp + (cbb + p1));
      acc0 = __builtin_amdgcn_wmma_f32_16x16x4_f32(false, A, false, B0, (short)0, acc0, false, false);
      acc1 = __builtin_amdgcn_wmma_f32_16x16x4_f32(false, A, false, B1, (short)0, acc1, false, false);
    }
  }

  // ---------------- Phase 3: normalize & scatter ---------------------------
  const int cA  = (wv    ) * 16 + lanelo;
  const int cB  = (wv + 8) * 16 + lanelo;
  const int mhi = (lane >= 16) ? 8 : 0;
  #pragma unroll
  for (int e = 0; e < 8; ++e) {
    const int M   = e + mhi;
    const int bin = mtile * 16 + M;
    if (bin < NBINS) {
      const float cnt = sCnt[M];
      const int ph = bin / POOLED;
      const int pw = bin % POOLED;
      const float o0 = (cnt > 0.f) ? (acc0[e] / cnt) : 0.f;
      const float o1 = (cnt > 0.f) ? (acc1[e] / cnt) : 0.f;
      out[((n*CH + cA)*POOLED + ph)*POOLED + pw] = o0;
      out[((n*CH + cB)*POOLED + ph)*POOLED + pw] = o1;
    }
  }
}

extern "C" void kernel_launch(void* const* d_in, const int* in_sizes, int n_in,
                              void* d_out, int out_size, void* d_ws, size_t ws_size,
                              hipStream_t stream) {
  (void)n_in; (void)out_size;
  const float* inp  = (const float*)d_in[0];
  const float* rois = (const float*)d_in[1];
  const float* offs = (const float*)d_in[2];
  float* out = (float*)d_out;

  const int B    = in_sizes[0] / (CH * HWSZ);   // 2
  const int nroi = in_sizes[1] / 5;             // 128
  const dim3 grid(4, nroi);                     // 4 bin-tiles x N rois

  const size_t need = (size_t)B * CH * HWSZ * sizeof(float);
  if (ws_size >= need) {
    // One-time layout transform: gathers become lane-contiguous (64B/half-wave).
    nchw_to_nhwc<<<dim3(HWSZ / 32, CH / 32, B), 256, 0, stream>>>(inp, (float*)d_ws);
    dcn_psroi_wmma<CH, 1><<<grid, 256, 0, stream>>>((const float*)d_ws, rois, offs, out);
  } else {
    dcn_psroi_wmma<1, HWSZ><<<grid, 256, 0, stream>>>(inp, rois, offs, out);
  }
}